// KAN_30193620091063
// MI455X (gfx1250) — compile-verified
//
#include <hip/hip_runtime.h>
#include <hip/hip_bf16.h>
#include <math.h>

typedef __attribute__((ext_vector_type(2))) float v2f;
typedef __attribute__((ext_vector_type(8))) float v8f;

#define BATCH 16384
#define DDIM 64
#define QN 32
#define HN 8
#define ROWS_PER_BLOCK 32
#define HPAD 65   // padded LDS row stride (floats): lane*65+d -> conflict-free banks

__global__ void kan_zero_kernel(float* __restrict__ out, int n) {
    int i = blockIdx.x * blockDim.x + threadIdx.x;
    if (i < n) out[i] = 0.0f;
}

__global__ __launch_bounds__(256)
void kan_main_kernel(const float* __restrict__ h,
                     const float* __restrict__ W1,
                     const float* __restrict__ b1,
                     const float* __restrict__ W2,
                     const float* __restrict__ b2,
                     const float* __restrict__ av,
                     const float* __restrict__ bv,
                     const float* __restrict__ cv,
                     float* __restrict__ out) {
    __shared__ float hbuf[ROWS_PER_BLOCK * HPAD];
    __shared__ float b2s[QN];

    const int tid  = threadIdx.x;
    const int lane = tid & 31;
    const int wave = tid >> 5;
    const int rowbase = blockIdx.x * ROWS_PER_BLOCK;

    // ---- Stage h tile (32 rows x 64 cols) into LDS via async copy ----
    // element e = j*256 + tid : row m = e>>6, col dc = e&63 ; coalesced global reads.
    #pragma unroll
    for (int j = 0; j < 8; ++j) {
        int e  = j * 256 + tid;
        int m  = e >> 6;
        int dc = e & 63;
        unsigned long long ga =
            (unsigned long long)(const void*)(h + (unsigned long long)(rowbase + m) * DDIM + dc);
        unsigned lo = (unsigned)(unsigned long long)(const void*)(&hbuf[m * HPAD + dc]);
        asm volatile("global_load_async_to_lds_b32 %0, %1, off" :: "v"(lo), "v"(ga) : "memory");
    }

    // ---- Wave 0: b2 row sums via V_WMMA_F32_16X16X4_F32 (overlaps async copy) ----
    // C[m,n] += sum_k A[m,k]*1.0 ; A[m,k] = b2[(t*16+m)*64 + kc + k]
    if (wave == 0) {
        #pragma unroll
        for (int t = 0; t < 2; ++t) {
            const int m    = lane & 15;
            const int koff = (lane >> 4) * 2;     // lanes 0-15: K=kc+0/1, lanes 16-31: K=kc+2/3
            v8f c = {};
            v2f bmat = {1.0f, 1.0f};              // all-ones B (layout-insensitive)
            #pragma unroll
            for (int kc = 0; kc < DDIM; kc += 4) {
                v2f amat = *(const v2f*)(b2 + (t * 16 + m) * DDIM + kc + koff);
                c = __builtin_amdgcn_wmma_f32_16x16x4_f32(
                        false, amat, false, bmat, (short)0, c, false, false);
            }
            // C layout: lane 0 VGPR r = C[r,n] (m=0..7) ; lane 16 VGPR r = C[8+r,n]
            if (lane == 0) {
                #pragma unroll
                for (int r = 0; r < 8; ++r) b2s[t * 16 + r] = c[r];
            } else if (lane == 16) {
                #pragma unroll
                for (int r = 0; r < 8; ++r) b2s[t * 16 + 8 + r] = c[r];
            }
        }
    }

    asm volatile("s_wait_asynccnt 0" ::: "memory");
    __syncthreads();

    // ---- Main compute: lane = batch row, wave handles 4 q values ----
    // Params are wave-uniform -> scalar cache (s_load); VALU reads SGPR operands.
    const int qg = wave * 4;
    float oacc = 0.0f;

    #pragma unroll 1
    for (int qq = 0; qq < 4; ++qq) {
        const int q = qg + qq;
        const float* __restrict__ w1p = W1 + q * DDIM * HN;
        const float* __restrict__ b1p = b1 + q * DDIM * HN;
        const float* __restrict__ w2p = W2 + q * DDIM * HN;

        float acc0 = 0.0f, acc1 = 0.0f;
        #pragma unroll 4
        for (int d = 0; d < DDIM; ++d) {
            const float x = hbuf[lane * HPAD + d];            // conflict-free ds_load
            #pragma unroll
            for (int hh = 0; hh < HN; hh += 2) {
                float y0 = fmaf(x, w1p[d * HN + hh],     b1p[d * HN + hh]);
                float y1 = fmaf(x, w1p[d * HN + hh + 1], b1p[d * HN + hh + 1]);
                y0 = fmaxf(y0, 0.0f);                         // v_max_num_f32
                y1 = fmaxf(y1, 0.0f);
                acc0 = fmaf(w2p[d * HN + hh],     y0, acc0);
                acc1 = fmaf(w2p[d * HN + hh + 1], y1, acc1);
            }
        }
        float s = acc0 + acc1 + b2s[q];                       // b2 row-sum from WMMA
        float z = fmaf(bv[q], s, cv[q]);
        float e = __expf(-2.0f * fabsf(z));                   // v_exp_f32
        float th = (1.0f - e) * __builtin_amdgcn_rcpf(1.0f + e);
        th = copysignf(th, z);
        oacc = fmaf(av[q], th, oacc);
    }

    atomicAdd(&out[rowbase + lane], oacc);                    // 8 waves combine per row
}

extern "C" void kernel_launch(void* const* d_in, const int* in_sizes, int n_in,
                              void* d_out, int out_size, void* d_ws, size_t ws_size,
                              hipStream_t stream) {
    const float* h  = (const float*)d_in[0];
    const float* W1 = (const float*)d_in[1];
    const float* b1 = (const float*)d_in[2];
    const float* W2 = (const float*)d_in[3];
    const float* b2 = (const float*)d_in[4];
    const float* a  = (const float*)d_in[5];
    const float* b  = (const float*)d_in[6];
    const float* c  = (const float*)d_in[7];
    float* out = (float*)d_out;

    kan_zero_kernel<<<(out_size + 255) / 256, 256, 0, stream>>>(out, out_size);
    kan_main_kernel<<<BATCH / ROWS_PER_BLOCK, 256, 0, stream>>>(
        h, W1, b1, W2, b2, a, b, c, out);
    (void)in_sizes; (void)n_in; (void)d_ws; (void)ws_size;
}